// CenterLoss_41429254537841
// MI455X (gfx1250) — compile-verified
//
#include <hip/hip_runtime.h>

// CenterLoss on gfx1250 (MI455X), wave32.
//
// loss = (1/N) * sum_n clamp(||x_n - c_{label_n}||^2, 1e-12, 1e12) + (C-1)*1e-12
// (the masked-matrix clamp in the reference turns every off-label zero into 1e-12).
//
// Memory-bound: ~8.6 MB streamed -> ~0.37 us at 23.3 TB/s. Squared distances for
// each 16-sample tile are computed as diag(diff * diff^T) with chained
// V_WMMA_F32_16X16X4_F32 (K=128 in 32 steps). Because the f32 A-fragment
// (16x4: lane = M + 16*(K>=2), VGPR = K%2) and the B-fragment of A^T
// (4x16: lane = N + 16*(K>=2), VGPR = K%2) are per-lane identical, one LDS
// v2f load feeds both WMMA operands.

typedef float v2f __attribute__((ext_vector_type(2)));
typedef float v8f __attribute__((ext_vector_type(8)));

#define WAVES_PER_BLOCK 4
#define TILE_ROWS 16
#define DIM 128
#define LDS_STRIDE 132   // 128 + 4 pad: 32-lane v2f reads hit all 64 LDS banks once
#define CLAMP_MIN 1e-12f
#define CLAMP_MAX 1e12f

__global__ __launch_bounds__(WAVES_PER_BLOCK * 32)
void center_loss_tiles(const float* __restrict__ x,
                       const float* __restrict__ centers,
                       const int*   __restrict__ labels,
                       float*       __restrict__ partials,
                       int N)
{
    __shared__ float diff[WAVES_PER_BLOCK][TILE_ROWS * LDS_STRIDE];
    __shared__ float wave_sum[WAVES_PER_BLOCK];

    const int lane = threadIdx.x & 31;
    const int wave = threadIdx.x >> 5;
    const int tile = blockIdx.x * WAVES_PER_BLOCK + wave;   // one 16-row tile per wave
    float* ld = diff[wave];

    // ---- stage diff = x[row] - centers[label[row]] into LDS (coalesced f32x4) ----
    {
        const int r    = lane >> 1;       // row 0..15
        const int half = lane & 1;        // columns 0..63 / 64..127
        int row_g = tile * TILE_ROWS + r;
        row_g = (row_g < N) ? row_g : (N - 1);          // keep EXEC full; masked later
        const int lab = labels[row_g];
        const float4* xr = (const float4*)(x       + (size_t)row_g * DIM) + half * 16;
        const float4* cr = (const float4*)(centers + (size_t)lab   * DIM) + half * 16;
        float* dst = ld + r * LDS_STRIDE + half * 64;
#pragma unroll
        for (int j = 0; j < 16; ++j) {
            float4 xv = xr[j];
            float4 cv = cr[j];
            float4 dv;
            dv.x = xv.x - cv.x; dv.y = xv.y - cv.y;
            dv.z = xv.z - cv.z; dv.w = xv.w - cv.w;
            *(float4*)(dst + 4 * j) = dv;
        }
    }
    // Each wave only touches its own LDS slice; same-wave DS ordering is handled
    // by the compiler's s_wait_dscnt insertion — no __syncthreads needed here.

    // ---- G = diff * diff^T via V_WMMA_F32_16X16X4_F32, K = 128 in 32 chunks ----
    v8f acc = {};
    {
        const int m  = lane & 15;
        const int hi = lane >> 4;                         // K sub-pair select
        const float* arow = ld + m * LDS_STRIDE + 2 * hi;
#pragma unroll
        for (int kk = 0; kk < 32; ++kk) {
            v2f a = *(const v2f*)(arow + 4 * kk);
            // 8 args: (neg_a, A, neg_b, B, c_mod, C, reuse_a, reuse_b)
            acc = __builtin_amdgcn_wmma_f32_16x16x4_f32(
                false, a, false, a, (short)0, acc, false, false);
        }
    }

    // ---- diagonal of G = squared distances ----
    // D layout: VGPR j: lanes 0-15 -> (M=j, N=lane); lanes 16-31 -> (M=j+8, N=lane-16)
    // diag(m): m<8 -> lane m, component m; m>=8 -> lane m+16, component m-8.
    float v = 0.0f;
    {
        int dm = -1, m = 0;
        if (lane < 8)        { dm = lane;      m = lane;      }
        else if (lane >= 24) { dm = lane - 24; m = lane - 16; }
        if (dm >= 0) {
            float d = acc[0];
#pragma unroll
            for (int i = 1; i < 8; ++i) d = (dm == i) ? acc[i] : d;
            d = fminf(fmaxf(d, CLAMP_MIN), CLAMP_MAX);
            v = (tile * TILE_ROWS + m < N) ? d : 0.0f;
        }
    }

    // ---- deterministic reductions: wave shuffle tree, then fixed-order block sum ----
#pragma unroll
    for (int off = 16; off >= 1; off >>= 1)
        v += __shfl_xor(v, off, 32);
    if (lane == 0) wave_sum[wave] = v;
    __syncthreads();
    if (threadIdx.x == 0) {
        float s = 0.0f;
#pragma unroll
        for (int w = 0; w < WAVES_PER_BLOCK; ++w) s += wave_sum[w];
        partials[blockIdx.x] = s;
    }
}

__global__ __launch_bounds__(256)
void center_loss_finalize(const float* __restrict__ partials,
                          float*       __restrict__ out,
                          int nparts, int N, int C)
{
    __shared__ float sh[256];
    float v = 0.0f;
    for (int i = threadIdx.x; i < nparts; i += 256) v += partials[i];   // fixed order
    sh[threadIdx.x] = v;
    __syncthreads();
    for (int s = 128; s >= 1; s >>= 1) {
        if ((int)threadIdx.x < s) sh[threadIdx.x] += sh[threadIdx.x + s];
        __syncthreads();
    }
    if (threadIdx.x == 0)
        out[0] = sh[0] / (float)N + (float)(C - 1) * CLAMP_MIN;
}

extern "C" void kernel_launch(void* const* d_in, const int* in_sizes, int n_in,
                              void* d_out, int out_size, void* d_ws, size_t ws_size,
                              hipStream_t stream)
{
    const float* x       = (const float*)d_in[0];   // [N, D] f32
    const float* centers = (const float*)d_in[1];   // [C, D] f32
    const int*   labels  = (const int*)  d_in[2];   // [N] int

    const int N = in_sizes[2];
    const int D = in_sizes[0] / N;                  // expected 128
    const int C = in_sizes[1] / D;

    const int ntiles  = (N + TILE_ROWS - 1) / TILE_ROWS;
    const int nblocks = (ntiles + WAVES_PER_BLOCK - 1) / WAVES_PER_BLOCK;

    float* partials = (float*)d_ws;                 // nblocks floats of scratch

    center_loss_tiles<<<nblocks, WAVES_PER_BLOCK * 32, 0, stream>>>(
        x, centers, labels, partials, N);
    center_loss_finalize<<<1, 256, 0, stream>>>(
        partials, (float*)d_out, nblocks, N, C);
}